// GCNNet_18777597018266
// MI455X (gfx1250) — compile-verified
//
#include <hip/hip_runtime.h>
#include <hip/hip_bf16.h>
#include <math.h>

typedef __attribute__((ext_vector_type(2))) float v2f;
typedef __attribute__((ext_vector_type(8))) float v8f;

#define D_IN 512
#define D_HID 16
#define N_CLS 40

// ---------------------------------------------------------------------------
// K0: zero fill
// ---------------------------------------------------------------------------
__global__ void zero_f32(float* __restrict__ p, long long n) {
    long long i = (long long)blockIdx.x * blockDim.x + threadIdx.x;
    if (i < n) p[i] = 0.0f;
}

// ---------------------------------------------------------------------------
// K1: h1 = X @ W1 using V_WMMA_F32_16X16X4_F32.
// One wave computes a 16(node) x 16(hidden) tile; K=512 in steps of 4.
// W1 (512x16 = 32KB) staged in LDS, broadcast to all 8 waves of the block.
// ---------------------------------------------------------------------------
__global__ __launch_bounds__(256) void gemm1_wmma(const float* __restrict__ x,
                                                  const float* __restrict__ W1,
                                                  float* __restrict__ h1,
                                                  int n_nodes) {
    __shared__ float sW[D_IN * D_HID];   // 32 KB of 320 KB LDS
    for (int i = threadIdx.x; i < D_IN * D_HID; i += blockDim.x) sW[i] = W1[i];
    __syncthreads();

    const int wave = threadIdx.x >> 5;          // 0..7
    const int lane = threadIdx.x & 31;
    const int tile = blockIdx.x * 8 + wave;     // 16-node tile id
    const int row0 = tile * 16;
    if (row0 >= n_nodes) return;                // uniform per-wave guard (EXEC stays full)

    const int mrow  = lane & 15;                // A-matrix row within tile
    const int khalf = (lane >> 4) << 1;         // lanes 0-15 hold K{0,1}, 16-31 hold K{2,3}
    int arow_idx = row0 + mrow;
    if (arow_idx >= n_nodes) arow_idx = n_nodes - 1;   // clamp (tail safety; N%16==0 here)
    const float* arow = x + (size_t)arow_idx * D_IN + khalf;

    v8f c = {};
    for (int k = 0; k < D_IN; k += 4) {
        v2f a, b;
        a.x = arow[k];
        a.y = arow[k + 1];
        const int kb = k + khalf;
        b.x = sW[kb * D_HID + mrow];            // B[k+khalf][n],   n = lane&15
        b.y = sW[(kb + 1) * D_HID + mrow];      // B[k+khalf+1][n]
        // 8 args: (neg_a, A, neg_b, B, c_mod, C, reuse_a, reuse_b)
        c = __builtin_amdgcn_wmma_f32_16x16x4_f32(
                false, a, false, b, (short)0, c, false, false);
    }

    // D layout: VGPR j -> M = j + (lane>=16 ? 8 : 0), N = lane&15
    const int ncol  = lane & 15;
    const int mbase = row0 + ((lane >> 4) << 3);
#pragma unroll
    for (int j = 0; j < 8; ++j) {
        int r = mbase + j;
        if (r < n_nodes) h1[(size_t)r * D_HID + ncol] = c[j];
    }
}

// ---------------------------------------------------------------------------
// K2: SpMM layer 1: h2[dst][j] += val * h1[src][j].
// Thread = (edge, feature); 16 consecutive lanes read one h1 row coalesced.
// Accumulation via L2-resident f32 atomics (non-returning).
// ---------------------------------------------------------------------------
__global__ void spmm16(const int* __restrict__ ei, const float* __restrict__ ev,
                       const float* __restrict__ hin, float* __restrict__ hout,
                       int n_edges) {
    long long t = (long long)blockIdx.x * blockDim.x + threadIdx.x;
    int e = (int)(t >> 4);
    int j = (int)(t & 15);
    if (e >= n_edges) return;
    int src = ei[e];
    int dst = ei[(size_t)n_edges + e];
    float v = ev[e];
    atomicAdd(&hout[(size_t)dst * D_HID + j], v * hin[(size_t)src * D_HID + j]);
}

// ---------------------------------------------------------------------------
// K3: hw2 = relu(h2 + b1) @ W2   (fused bias + ReLU + tiny GEMM)
// ---------------------------------------------------------------------------
__global__ void gemm2_relu(const float* __restrict__ h2, const float* __restrict__ b1,
                           const float* __restrict__ W2, float* __restrict__ hw2,
                           int n_nodes) {
    long long t = (long long)blockIdx.x * blockDim.x + threadIdx.x;
    int n = (int)(t / N_CLS);
    int c = (int)(t % N_CLS);
    if (n >= n_nodes) return;
    float acc = 0.0f;
#pragma unroll
    for (int k = 0; k < D_HID; ++k) {
        float hk = h2[(size_t)n * D_HID + k] + b1[k];
        hk = hk > 0.0f ? hk : 0.0f;
        acc = fmaf(hk, W2[k * N_CLS + c], acc);
    }
    hw2[(size_t)n * N_CLS + c] = acc;
}

// ---------------------------------------------------------------------------
// K4: SpMM layer 2: out[dst][c] += val * hw2[src][c]  (40 classes)
// ---------------------------------------------------------------------------
__global__ void spmm40(const int* __restrict__ ei, const float* __restrict__ ev,
                       const float* __restrict__ hin, float* __restrict__ out,
                       int n_edges) {
    long long t = (long long)blockIdx.x * blockDim.x + threadIdx.x;
    long long total = (long long)n_edges * N_CLS;
    if (t >= total) return;
    int e = (int)(t / N_CLS);
    int c = (int)(t % N_CLS);
    int src = ei[e];
    int dst = ei[(size_t)n_edges + e];
    float v = ev[e];
    atomicAdd(&out[(size_t)dst * N_CLS + c], v * hin[(size_t)src * N_CLS + c]);
}

// ---------------------------------------------------------------------------
// K5: in-place  out = log_softmax(out + b2)  per row of 40
// ---------------------------------------------------------------------------
__global__ void logsoftmax40(float* __restrict__ logits, const float* __restrict__ b2,
                             int n_nodes) {
    int n = blockIdx.x * blockDim.x + threadIdx.x;
    if (n >= n_nodes) return;
    float* row = logits + (size_t)n * N_CLS;
    float vals[N_CLS];
    float m = -INFINITY;
#pragma unroll
    for (int c = 0; c < N_CLS; ++c) {
        vals[c] = row[c] + b2[c];
        m = fmaxf(m, vals[c]);
    }
    float s = 0.0f;
#pragma unroll
    for (int c = 0; c < N_CLS; ++c) s += __expf(vals[c] - m);
    float lse = m + __logf(s);
#pragma unroll
    for (int c = 0; c < N_CLS; ++c) row[c] = vals[c] - lse;
}

// ---------------------------------------------------------------------------
// Launcher
// ---------------------------------------------------------------------------
extern "C" void kernel_launch(void* const* d_in, const int* in_sizes, int n_in,
                              void* d_out, int out_size, void* d_ws, size_t ws_size,
                              hipStream_t stream) {
    const float* x  = (const float*)d_in[0];
    const int*   ei = (const int*)d_in[1];
    const float* ev = (const float*)d_in[2];
    const float* W1 = (const float*)d_in[3];
    const float* b1 = (const float*)d_in[4];
    const float* W2 = (const float*)d_in[5];
    const float* b2 = (const float*)d_in[6];
    float* out = (float*)d_out;

    const int N = in_sizes[0] / D_IN;   // 100000
    const int E = in_sizes[2];          // 3200000

    float* h1  = (float*)d_ws;                         // [N,16]
    float* h2  = h1 + (size_t)N * D_HID;               // [N,16]
    float* hw2 = h2 + (size_t)N * D_HID;               // [N,40]

    // zero the atomic accumulators (h2 and d_out)
    long long nz = (long long)N * D_HID;
    zero_f32<<<(unsigned)((nz + 255) / 256), 256, 0, stream>>>(h2, nz);
    long long no = (long long)N * N_CLS;
    zero_f32<<<(unsigned)((no + 255) / 256), 256, 0, stream>>>(out, no);

    // GEMM1 (WMMA): 8 tiles of 16 nodes per 256-thread block
    int tiles  = (N + 15) / 16;
    int blocks = (tiles + 7) / 8;
    gemm1_wmma<<<blocks, 256, 0, stream>>>(x, W1, h1, N);

    // SpMM1
    long long t1 = (long long)E * D_HID;
    spmm16<<<(unsigned)((t1 + 255) / 256), 256, 0, stream>>>(ei, ev, h1, h2, E);

    // bias + ReLU + GEMM2
    long long t2 = (long long)N * N_CLS;
    gemm2_relu<<<(unsigned)((t2 + 255) / 256), 256, 0, stream>>>(h2, b1, W2, hw2, N);

    // SpMM2 into d_out
    long long t3 = (long long)E * N_CLS;
    spmm40<<<(unsigned)((t3 + 255) / 256), 256, 0, stream>>>(ei, ev, hw2, out, E);

    // bias + log_softmax in place
    logsoftmax40<<<(N + 255) / 256, 256, 0, stream>>>(out, b2, N);
}